// GNN_24404004176133
// MI455X (gfx1250) — compile-verified
//
#include <hip/hip_runtime.h>

// ---------------------------------------------------------------------------
// GraphSAGE (gcn aggregator) x4 + mean pooling, for MI455X (gfx1250, wave32).
// Dense per-node GEMMs use V_WMMA_F32_16X16X4_F32 (f32 WMMA, 16x16 C tiles).
// Irregular gather/scatter uses global f32 atomics (L2-resident working set).
// N_NODES = 100000 is exactly 6250 16-row tiles -> all valid tiles are full,
// so the WMMA kernel needs only one wave-uniform range check (no exec ladders).
// ---------------------------------------------------------------------------

typedef float v2f __attribute__((ext_vector_type(2)));
typedef float v8f __attribute__((ext_vector_type(8)));

#define N_NODES_C  100000
#define N_EDGES_C  1600000
#define N_GRAPHS_C 64
#define EMB_C      64

// ---------------------------------------------------------------- zero fill
__global__ void gnn_zero_f32(float* __restrict__ p, int n) {
  int t = blockIdx.x * blockDim.x + threadIdx.x;
  int stride = gridDim.x * blockDim.x;
  for (; t < n; t += stride) p[t] = 0.0f;
}

// ---------------------------------------------------------------- in-degree
__global__ void gnn_degree(const int* __restrict__ dst, float* __restrict__ deg,
                           int nEdges) {
  int e = blockIdx.x * blockDim.x + threadIdx.x;
  if (e < nEdges) atomicAdd(&deg[dst[e]], 1.0f);
}

// deg[i] -> 1 / (deg[i] + 1), computed once, reused by all 4 layers.
__global__ void gnn_inv_deg(float* __restrict__ deg, int n) {
  int t = blockIdx.x * blockDim.x + threadIdx.x;
  if (t < n) deg[t] = 1.0f / (deg[t] + 1.0f);
}

// --------------------------------------------------- edge gather/scatter-add
// neigh[dst[e]][f] += h[src[e]][f], one thread per (edge, feature).
template <int DIN>
__global__ void gnn_scatter_add(const float* __restrict__ h,
                                const int* __restrict__ src,
                                const int* __restrict__ dst,
                                float* __restrict__ neigh, int nEdges) {
  constexpr bool POW2 = (DIN & (DIN - 1)) == 0;
  constexpr int  SH   = (DIN == 16) ? 4 : (DIN == 32) ? 5 : 6;  // used if POW2
  int t = blockIdx.x * blockDim.x + threadIdx.x;
  const int total  = nEdges * DIN;
  const int stride = gridDim.x * blockDim.x;
  for (; t < total; t += stride) {
    int e, f;
    if constexpr (POW2) {
      e = t >> SH;
      f = t & (DIN - 1);
    } else {
      e = t / DIN;
      f = t - e * DIN;
    }
    // pull edge indices ahead into cache (global_prefetch_b8 on gfx1250)
    __builtin_prefetch(&src[e + 2048], 0, 0);
    const int s = src[e];
    const int d = dst[e];
    atomicAdd(&neigh[d * DIN + f], h[s * DIN + f]);
  }
}

// ---------------------------------------------- normalize + GEMM + bias+ReLU
// hout = relu( ((neigh + hin) * inv) @ W + b )     via v_wmma_f32_16x16x4_f32
// One wave per full 16-node tile; DOUT/16 column tiles; K padded to mult of 4.
// W is staged zero-padded into LDS by the whole block (removes lane-varying
// guards from the inner loop); A rows use a padded LDS stride to avoid 16-way
// bank conflicts on fragment loads.
template <int DIN, int DOUT>
__global__ __launch_bounds__(256) void gnn_sage_wmma(
    const float* __restrict__ hin, const float* __restrict__ neigh,
    const float* __restrict__ inv, const float* __restrict__ W,
    const float* __restrict__ bias, float* __restrict__ hout, int nTiles) {
  constexpr int KPAD  = (DIN + 3) & ~3;
  constexpr int LDA   = KPAD + 4;  // padded row stride (banks), 8B-aligned
  constexpr int KK    = KPAD / 4;
  constexpr int NT    = DOUT / 16;
  constexpr int WAVES = 256 / 32;

  __shared__ float tileA[WAVES][16 * LDA];
  __shared__ float wl[KPAD * DOUT];

  // ---- Block-cooperative: stage zero-padded W into LDS (KPAD x DOUT).
  for (int idx = threadIdx.x; idx < KPAD * DOUT; idx += 256) {
    const int k = idx / DOUT;          // DOUT is a power of two -> shifts
    const int n = idx - k * DOUT;
    wl[idx] = (k < DIN) ? W[k * DOUT + n] : 0.0f;
  }
  __syncthreads();  // every wave reaches this before any early-out

  const int wave = threadIdx.x >> 5;
  const int lane = threadIdx.x & 31;
  const int tile = blockIdx.x * WAVES + wave;
  if (tile >= nTiles) return;          // wave-uniform: full EXEC inside
  const int row0 = tile * 16;

  float* tl = tileA[wave];

  // ---- Stage normalized A tile (16 x KPAD) into per-wave LDS.
  for (int idx = lane; idx < 16 * KPAD; idx += 32) {
    const int r = idx / KPAD;
    const int c = idx - r * KPAD;
    const int node = row0 + r;
    float v = 0.0f;
    if (DIN == KPAD || c < DIN)
      v = (neigh[node * DIN + c] + hin[node * DIN + c]) * inv[node];
    tl[r * LDA + c] = v;
  }
  // Same-wave LDS ops retire in order (DScnt); no extra barrier needed.

  const int rowA = lane & 15;          // A rows: M = lane%16
  const int kSel = (lane >> 4) << 1;   // lanes 0-15 -> K+{0,1}; 16-31 -> +{2,3}
  const int col  = lane & 15;          // B/C/D column: N = lane%16

  v8f acc[NT] = {};

#pragma unroll
  for (int kk = 0; kk < KK; ++kk) {
    const int k0 = kk * 4 + kSel;
    v2f a;
    a.x = tl[rowA * LDA + k0];
    a.y = tl[rowA * LDA + k0 + 1];
#pragma unroll
    for (int nt = 0; nt < NT; ++nt) {
      const int n = nt * 16 + col;
      v2f bm;
      bm.x = wl[k0 * DOUT + n];
      bm.y = wl[(k0 + 1) * DOUT + n];
      acc[nt] = __builtin_amdgcn_wmma_f32_16x16x4_f32(
          /*neg_a=*/false, a, /*neg_b=*/false, bm,
          /*c_mod=*/(short)0, acc[nt], /*reuse_a=*/false, /*reuse_b=*/false);
    }
  }

  // ---- Epilogue: D VGPR v -> row M = v + (lane>=16 ? 8 : 0), col = lane%16.
  const int rowBase = (lane >> 4) << 3;
#pragma unroll
  for (int nt = 0; nt < NT; ++nt) {
    const int n = nt * 16 + col;
    const float bv = bias[n];
#pragma unroll
    for (int v = 0; v < 8; ++v) {
      const int node = row0 + rowBase + v;
      const float x = acc[nt][v] + bv;
      hout[node * DOUT + n] = x > 0.0f ? x : 0.0f;
    }
  }
}

// ------------------------------------------------------- per-graph mean pool
__global__ void gnn_pool_sum(const float* __restrict__ h,
                             const int* __restrict__ gids,
                             float* __restrict__ sums, float* __restrict__ cnts,
                             int nNodes) {
  int t = blockIdx.x * blockDim.x + threadIdx.x;
  const int total = nNodes * EMB_C;
  if (t >= total) return;
  const int n = t >> 6;  // EMB_C == 64
  const int f = t & 63;
  const int g = gids[n];
  atomicAdd(&sums[g * EMB_C + f], h[t]);
  if (f == 0) atomicAdd(&cnts[g], 1.0f);
}

__global__ void gnn_pool_finalize(const float* __restrict__ sums,
                                  const float* __restrict__ cnts,
                                  float* __restrict__ out) {
  int t = blockIdx.x * blockDim.x + threadIdx.x;
  if (t < N_GRAPHS_C * EMB_C) {
    const float c = cnts[t >> 6];
    out[t] = sums[t] / fmaxf(c, 1.0f);
  }
}

// ---------------------------------------------------------------------------
extern "C" void kernel_launch(void* const* d_in, const int* in_sizes, int n_in,
                              void* d_out, int out_size, void* d_ws,
                              size_t ws_size, hipStream_t stream) {
  (void)in_sizes; (void)n_in; (void)out_size; (void)ws_size;

  const float* feat = (const float*)d_in[0];   // [100000, 6]
  const int*   src  = (const int*)d_in[1];     // [1.6M]
  const int*   dst  = (const int*)d_in[2];     // [1.6M]
  const int*   gids = (const int*)d_in[3];     // [100000]
  const float* W1 = (const float*)d_in[4];  const float* b1 = (const float*)d_in[5];
  const float* W2 = (const float*)d_in[6];  const float* b2 = (const float*)d_in[7];
  const float* W3 = (const float*)d_in[8];  const float* b3 = (const float*)d_in[9];
  const float* W4 = (const float*)d_in[10]; const float* b4 = (const float*)d_in[11];
  float* out = (float*)d_out;

  // Workspace layout (floats).
  float* ws    = (float*)d_ws;
  float* hA    = ws;                                  // [100000*64]
  float* hB    = hA + (size_t)N_NODES_C * EMB_C;      // [100000*64]
  float* neigh = hB + (size_t)N_NODES_C * EMB_C;      // [100000*64]
  float* inv   = neigh + (size_t)N_NODES_C * EMB_C;   // [100000] (deg -> inv)
  float* sums  = inv + N_NODES_C;                     // [64*64]
  float* cnts  = sums + N_GRAPHS_C * EMB_C;           // [64]

  const int ZB = 2048, TB = 256;
  const int tiles   = N_NODES_C / 16 + (N_NODES_C % 16 != 0);  // 6250 (exact)
  const int wblocks = (tiles + 7) / 8;                         // 8 waves/block

  // Degree -> 1/(deg+1), computed once and reused by all layers.
  gnn_zero_f32<<<ZB, TB, 0, stream>>>(inv, N_NODES_C);
  gnn_degree<<<(N_EDGES_C + TB - 1) / TB, TB, 0, stream>>>(dst, inv, N_EDGES_C);
  gnn_inv_deg<<<(N_NODES_C + TB - 1) / TB, TB, 0, stream>>>(inv, N_NODES_C);

  // ---- Layer 1: feat[.,6] -> hA[.,16]
  gnn_zero_f32<<<ZB, TB, 0, stream>>>(neigh, N_NODES_C * 6);
  gnn_scatter_add<6><<<20480, TB, 0, stream>>>(feat, src, dst, neigh, N_EDGES_C);
  gnn_sage_wmma<6, 16><<<wblocks, TB, 0, stream>>>(feat, neigh, inv, W1, b1, hA, tiles);

  // ---- Layer 2: hA[.,16] -> hB[.,32]
  gnn_zero_f32<<<ZB, TB, 0, stream>>>(neigh, N_NODES_C * 16);
  gnn_scatter_add<16><<<40960, TB, 0, stream>>>(hA, src, dst, neigh, N_EDGES_C);
  gnn_sage_wmma<16, 32><<<wblocks, TB, 0, stream>>>(hA, neigh, inv, W2, b2, hB, tiles);

  // ---- Layer 3: hB[.,32] -> hA[.,64]
  gnn_zero_f32<<<ZB, TB, 0, stream>>>(neigh, N_NODES_C * 32);
  gnn_scatter_add<32><<<40960, TB, 0, stream>>>(hB, src, dst, neigh, N_EDGES_C);
  gnn_sage_wmma<32, 64><<<wblocks, TB, 0, stream>>>(hB, neigh, inv, W3, b3, hA, tiles);

  // ---- Layer 4: hA[.,64] -> hB[.,64]
  gnn_zero_f32<<<ZB, TB, 0, stream>>>(neigh, N_NODES_C * 64);
  gnn_scatter_add<64><<<40960, TB, 0, stream>>>(hA, src, dst, neigh, N_EDGES_C);
  gnn_sage_wmma<64, 64><<<wblocks, TB, 0, stream>>>(hA, neigh, inv, W4, b4, hB, tiles);

  // ---- Per-graph mean pooling over hB (sums and cnts are contiguous).
  gnn_zero_f32<<<ZB, TB, 0, stream>>>(sums, N_GRAPHS_C * EMB_C + N_GRAPHS_C);
  gnn_pool_sum<<<(N_NODES_C * EMB_C + TB - 1) / TB, TB, 0, stream>>>(hB, gids, sums, cnts, N_NODES_C);
  gnn_pool_finalize<<<(N_GRAPHS_C * EMB_C + TB - 1) / TB, TB, 0, stream>>>(sums, cnts, out);
}